// ChannelAttentionModule_48387101557180
// MI455X (gfx1250) — compile-verified
//
#include <hip/hip_runtime.h>
#include <hip/hip_bf16.h>

typedef __attribute__((ext_vector_type(2))) float v2f;
typedef __attribute__((ext_vector_type(8))) float v8f;

#define NB    16        // batch
#define NC    128       // channels
#define HW    65536     // 256*256 spatial
#define SPLIT 4         // chunks per (b,c) row
#define CHUNK (HW / SPLIT)          // 16384 floats per block
#define TPB   256                   // 8 wave32s per block

// ---------------------------------------------------------------------------
// Stage 1: bandwidth-bound streaming reduction.
// Grid: NB*NC*SPLIT blocks; block (bc,part) sums CHUNK contiguous floats and
// writes one partial sum to sums[bc*SPLIT+part]. Each slot written exactly
// once -> no atomics, no workspace zeroing needed.
// ---------------------------------------------------------------------------
__global__ void __launch_bounds__(TPB)
pool_reduce_kernel(const float* __restrict__ x, float* __restrict__ sums) {
    const int bidx = blockIdx.x;            // [0, NB*NC*SPLIT)
    const int bc   = bidx >> 2;
    const int part = bidx & (SPLIT - 1);
    const float4* p =
        (const float4*)(x + (size_t)bc * HW + (size_t)part * CHUNK);
    const int t = threadIdx.x;

    float acc = 0.0f;
    // CHUNK / (TPB*4) = 16 B128 loads per thread, fully unrolled so the
    // compiler can keep many global_load_b128 in flight (latency hiding).
#pragma unroll
    for (int j = 0; j < CHUNK / (TPB * 4); ++j) {
        float4 v = p[j * TPB + t];
        acc += (v.x + v.y) + (v.z + v.w);
    }

    // wave32 butterfly reduction
#pragma unroll
    for (int off = 16; off > 0; off >>= 1)
        acc += __shfl_xor(acc, off, 32);

    __shared__ float wsum[TPB / 32];
    if ((t & 31) == 0) wsum[t >> 5] = acc;
    __syncthreads();

    if (t == 0) {
        float s = 0.0f;
#pragma unroll
        for (int i = 0; i < TPB / 32; ++i) s += wsum[i];
        sums[bidx] = s;
    }
}

// ---------------------------------------------------------------------------
// Stage 2: pooled[16,128] @ W[128] + b, sigmoid -> out[16].
// Single wave32; 32x V_WMMA_F32_16X16X4_F32 accumulating over K in steps of 4.
//   A (16x4 f32, 2 VGPRs): lanes 0-15 hold M=lane {K=k0,k0+1},
//                          lanes 16-31 hold M=lane-16 {K=k0+2,k0+3}.
//   B (4x16): row K broadcast across the 16 columns -> every D column equals
//             the desired dot product; we read column N=0.
//   D (16x16 f32, 8 VGPRs): VGPR r = M=r (lanes 0-15) / M=r+8 (lanes 16-31).
// EXEC is all-1s through every WMMA (divergence only after the loop).
// ---------------------------------------------------------------------------
__global__ void __launch_bounds__(32)
gemv_sigmoid_wmma_kernel(const float* __restrict__ sums,
                         const float* __restrict__ W,
                         const float* __restrict__ bias,
                         float* __restrict__ out) {
    const int lane = threadIdx.x;   // 0..31
    const int row  = lane & 15;     // M index this lane carries for A
    const int hi   = lane >> 4;     // 0: K=k0,k0+1   1: K=k0+2,k0+3
    const float scale = 1.0f / (float)HW;

    v8f c = {};
#pragma unroll 4
    for (int k0 = 0; k0 < NC; k0 += 4) {
        const int ka = k0 + hi * 2;
        const float* s0 = sums + (size_t)(row * NC + ka) * SPLIT;
        const float* s1 = sums + (size_t)(row * NC + ka + 1) * SPLIT;
        v2f a, b;
        a.x = (s0[0] + s0[1] + s0[2] + s0[3]) * scale;   // pooled[row][ka]
        a.y = (s1[0] + s1[1] + s1[2] + s1[3]) * scale;   // pooled[row][ka+1]
        b.x = W[ka];                                     // B row broadcast
        b.y = W[ka + 1];
        // (neg_a, A, neg_b, B, c_mod, C, reuse_a, reuse_b)
        c = __builtin_amdgcn_wmma_f32_16x16x4_f32(
                false, a, false, b, (short)0, c, false, false);
    }

    const float bv = bias[0];
    // Column N=0 lives in lane 0 (M=0..7) and lane 16 (M=8..15).
    if ((lane & 15) == 0) {
#pragma unroll
        for (int r = 0; r < 8; ++r) {
            float y = c[r] + bv;
            out[hi * 8 + r] = 1.0f / (1.0f + __expf(-y));
        }
    }
}

// ---------------------------------------------------------------------------
extern "C" void kernel_launch(void* const* d_in, const int* in_sizes, int n_in,
                              void* d_out, int out_size, void* d_ws, size_t ws_size,
                              hipStream_t stream) {
    const float* x    = (const float*)d_in[0];   // [16,128,256,256] fp32
    const float* W    = (const float*)d_in[1];   // [1,128] fp32
    const float* bias = (const float*)d_in[2];   // [1] fp32
    float* out  = (float*)d_out;                 // [16] fp32
    float* sums = (float*)d_ws;                  // NB*NC*SPLIT floats = 32 KiB

    pool_reduce_kernel<<<NB * NC * SPLIT, TPB, 0, stream>>>(x, sums);
    gemv_sigmoid_wmma_kernel<<<1, 32, 0, stream>>>(sums, W, bias, out);
}